// GATModel_29832842838750
// MI455X (gfx1250) — compile-verified
//
#include <hip/hip_runtime.h>

#define NTOT   16384
#define KNBR   8
#define HIDC   128
#define NCLS   3
#define SLOPE  0.2f
#define SPLIT  4
#define TILE   256

typedef __attribute__((ext_vector_type(2))) float v2f;
typedef __attribute__((ext_vector_type(8))) float v8f;

// ---------------------------------------------------------------------------
// Kernel 1: streaming kNN, pass 1. One thread per row, scans 1/SPLIT of the
// columns. Column tiles staged in LDS as packed float4 {x,y,z,batch} so the
// inner loop is a single broadcast ds_load_b128 + ~8 VALU ops per candidate.
// Keeps top-8 smallest distances in a sorted register file.
// ---------------------------------------------------------------------------
__global__ void knn_pass1_kernel(const float* __restrict__ coord,
                                 const int*   __restrict__ batch,
                                 float* __restrict__ cand_d,
                                 int*   __restrict__ cand_j) {
  __shared__ float4 tile[TILE];
  const int i = blockIdx.x * TILE + threadIdx.x;   // row
  const int s = blockIdx.y;                        // column chunk
  const float cx = coord[i * 3 + 0];
  const float cy = coord[i * 3 + 1];
  const float cz = coord[i * 3 + 2];
  const int   bi = batch[i];

  float bd[KNBR];
  int   bj[KNBR];
#pragma unroll
  for (int t = 0; t < KNBR; ++t) { bd[t] = 3.0e38f; bj[t] = 0; }

  const int jbeg = s * (NTOT / SPLIT);
  const int jend = jbeg + (NTOT / SPLIT);
  for (int j0 = jbeg; j0 < jend; j0 += TILE) {
    const int jl = j0 + threadIdx.x;
    float4 v;
    v.x = coord[jl * 3 + 0];
    v.y = coord[jl * 3 + 1];
    v.z = coord[jl * 3 + 2];
    v.w = __int_as_float(batch[jl]);
    if (j0 + TILE < jend)   // speculative prefetch of next tile (global_prefetch_b8)
      __builtin_prefetch(&coord[(jl + TILE) * 3], 0, 1);
    __syncthreads();
    tile[threadIdx.x] = v;
    __syncthreads();

#pragma unroll 4
    for (int jj = 0; jj < TILE; ++jj) {
      const float4 c = tile[jj];
      if (__float_as_int(c.w) != bi) continue;     // cross-graph pair masked
      const float dx = c.x - cx, dy = c.y - cy, dz = c.z - cz;
      const float d2 = dx * dx + dy * dy + dz * dz;
      if (d2 < bd[KNBR - 1]) {
        float dc = d2; int jc = j0 + jj;
#pragma unroll
        for (int t = 0; t < KNBR; ++t) {
          if (dc < bd[t]) {
            float td = bd[t]; int tj = bj[t];
            bd[t] = dc; bj[t] = jc; dc = td; jc = tj;
          }
        }
      }
    }
  }
#pragma unroll
  for (int t = 0; t < KNBR; ++t) {
    cand_d[(i * SPLIT + s) * KNBR + t] = bd[t];
    cand_j[(i * SPLIT + s) * KNBR + t] = bj[t];
  }
}

// ---------------------------------------------------------------------------
// Kernel 2: merge the SPLIT partial top-8 lists into the final top-8 indices.
// ---------------------------------------------------------------------------
__global__ void knn_merge_kernel(const float* __restrict__ cand_d,
                                 const int*   __restrict__ cand_j,
                                 int*   __restrict__ idx) {
  const int i = blockIdx.x * blockDim.x + threadIdx.x;
  float bd[KNBR];
  int   bj[KNBR];
#pragma unroll
  for (int t = 0; t < KNBR; ++t) { bd[t] = 3.0e38f; bj[t] = 0; }
  for (int c = 0; c < SPLIT * KNBR; ++c) {
    const float d2 = cand_d[i * SPLIT * KNBR + c];
    const int   j  = cand_j[i * SPLIT * KNBR + c];
    if (d2 < bd[KNBR - 1]) {
      float dc = d2; int jc = j;
#pragma unroll
      for (int t = 0; t < KNBR; ++t) {
        if (dc < bd[t]) {
          float td = bd[t]; int tj = bj[t];
          bd[t] = dc; bj[t] = jc; dc = td; jc = tj;
        }
      }
    }
  }
#pragma unroll
  for (int t = 0; t < KNBR; ++t) idx[i * KNBR + t] = bj[t];
}

// ---------------------------------------------------------------------------
// Kernel 3: Y = X @ W + b via V_WMMA_F32_16X16X4_F32 (full fp32 matrix pipe).
// DIN is a compile-time constant so the K-loop fully unrolls with NO exec
// masking: for DIN=128 the body is pure load_b64/load_b32 + v_wmma; for
// DIN=6 the 2-wide tail uses clamped-address loads + v_cndmask selects.
// Block = 256 threads = 8 waves; block covers a 16-row strip, wave w covers
// column tile [16w, 16w+16).
// A frag (16x4, 2 VGPR): lanes 0-15 -> M=lane, K={0,1}; lanes 16-31 -> K={2,3}.
// C/D (16x16, 8 VGPR):  vgpr v -> M = v + (lane<16 ? 0 : 8), N = lane&15.
// ---------------------------------------------------------------------------
template <int DIN>
__global__ void gemm_bias_wmma_kernel(const float* __restrict__ X,
                                      const float* __restrict__ W,
                                      const float* __restrict__ bias,
                                      float* __restrict__ Y) {
  const int lane = threadIdx.x & 31;
  const int wave = threadIdx.x >> 5;
  const int row0 = blockIdx.x * 16;
  const int col0 = wave * 16;
  const int m    = lane & 15;
  const int koff = (lane >> 4) << 1;   // 0 for lanes 0-15, 2 for lanes 16-31

  const float* __restrict__ Xrow = X + (size_t)(row0 + m) * DIN;
  const float* __restrict__ Wcol = W + col0 + m;

  constexpr int KPAD = (DIN + 3) & ~3;
  v8f c = {0.f, 0.f, 0.f, 0.f, 0.f, 0.f, 0.f, 0.f};
#pragma unroll
  for (int k0 = 0; k0 < KPAD; k0 += 4) {
    const int ka = k0 + koff;
    v2f a, b;
    if constexpr ((DIN & 3) == 0) {
      // no tail: unconditional b64 (A) and 2x b32 (W) loads
      a.x = Xrow[ka];
      a.y = Xrow[ka + 1];
      b.x = Wcol[(size_t)ka * HIDC];
      b.y = Wcol[(size_t)(ka + 1) * HIDC];
    } else {
      // tail handled branch-free: clamp address, select 0 via v_cndmask
      const bool ok0 = (ka     < DIN);
      const bool ok1 = (ka + 1 < DIN);
      const int  k0c = ok0 ? ka     : 0;
      const int  k1c = ok1 ? ka + 1 : 0;
      const float ax = Xrow[k0c];
      const float ay = Xrow[k1c];
      const float bx = Wcol[(size_t)k0c * HIDC];
      const float by = Wcol[(size_t)k1c * HIDC];
      a.x = ok0 ? ax : 0.f;
      a.y = ok1 ? ay : 0.f;
      b.x = ok0 ? bx : 0.f;
      b.y = ok1 ? by : 0.f;
    }
    c = __builtin_amdgcn_wmma_f32_16x16x4_f32(false, a, false, b,
                                              (short)0, c, false, false);
  }
  const float bv  = bias[col0 + m];
  const int   mhi = (lane >> 4) * 8;
#pragma unroll
  for (int v = 0; v < 8; ++v) {
    Y[(size_t)(row0 + mhi + v) * HIDC + col0 + m] = c[v] + bv;
  }
}

// ---------------------------------------------------------------------------
// Kernel 4: GATv2 attention + aggregation + bias + ReLU, fused.
// One wave per node; lane owns 4 contiguous channels (float4 path).
// logit_k = sum_h att[h] * leaky_relu(xl[j_k,h] + xr[i,h])  (wave reduction)
// out[i]  = relu( sum_k softmax(logit)_k * xl[j_k] + bias )
// ---------------------------------------------------------------------------
__global__ void gat_aggregate_kernel(const float* __restrict__ xl,
                                     const float* __restrict__ xr,
                                     const int*   __restrict__ idx,
                                     const float* __restrict__ att,
                                     const float* __restrict__ bias,
                                     float* __restrict__ out) {
  const int lane = threadIdx.x & 31;
  const int wave = threadIdx.x >> 5;
  const int i    = blockIdx.x * 8 + wave;
  const int h0   = lane * 4;           // HIDC = 32 lanes * 4 channels

  const float4 xr4 = *(const float4*)&xr[i * HIDC + h0];
  const float4 at4 = *(const float4*)&att[h0];

  float4 g[KNBR];
  float  logit[KNBR];
#pragma unroll
  for (int k = 0; k < KNBR; ++k) {
    const int j = idx[i * KNBR + k];
    g[k] = *(const float4*)&xl[j * HIDC + h0];
    float ex = g[k].x + xr4.x; ex = (ex > 0.f) ? ex : SLOPE * ex;
    float ey = g[k].y + xr4.y; ey = (ey > 0.f) ? ey : SLOPE * ey;
    float ez = g[k].z + xr4.z; ez = (ez > 0.f) ? ez : SLOPE * ez;
    float ew = g[k].w + xr4.w; ew = (ew > 0.f) ? ew : SLOPE * ew;
    float p = ex * at4.x + ey * at4.y + ez * at4.z + ew * at4.w;
#pragma unroll
    for (int off = 16; off > 0; off >>= 1) p += __shfl_xor(p, off, 32);
    logit[k] = p;
  }
  float mx = logit[0];
#pragma unroll
  for (int k = 1; k < KNBR; ++k) mx = fmaxf(mx, logit[k]);
  float w[KNBR], sum = 0.f;
#pragma unroll
  for (int k = 0; k < KNBR; ++k) { w[k] = __expf(logit[k] - mx); sum += w[k]; }
  const float inv = 1.f / sum;

  float4 acc = {0.f, 0.f, 0.f, 0.f};
#pragma unroll
  for (int k = 0; k < KNBR; ++k) {
    const float a = w[k] * inv;
    acc.x += a * g[k].x; acc.y += a * g[k].y;
    acc.z += a * g[k].z; acc.w += a * g[k].w;
  }
  const float4 b4 = *(const float4*)&bias[h0];
  acc.x = fmaxf(acc.x + b4.x, 0.f);
  acc.y = fmaxf(acc.y + b4.y, 0.f);
  acc.z = fmaxf(acc.z + b4.z, 0.f);
  acc.w = fmaxf(acc.w + b4.w, 0.f);
  *(float4*)&out[i * HIDC + h0] = acc;
}

// ---------------------------------------------------------------------------
// Kernel 5: final classifier head, out = x @ Wf + bf  (12 MFLOP, VALU).
// ---------------------------------------------------------------------------
__global__ void final_linear_kernel(const float* __restrict__ x,
                                    const float* __restrict__ Wf,
                                    const float* __restrict__ bf,
                                    float* __restrict__ out) {
  const int i = blockIdx.x * blockDim.x + threadIdx.x;
  float a0 = bf[0], a1 = bf[1], a2 = bf[2];
  for (int h = 0; h < HIDC; ++h) {
    const float xv = x[i * HIDC + h];
    a0 += xv * Wf[h * NCLS + 0];
    a1 += xv * Wf[h * NCLS + 1];
    a2 += xv * Wf[h * NCLS + 2];
  }
  out[i * NCLS + 0] = a0;
  out[i * NCLS + 1] = a1;
  out[i * NCLS + 2] = a2;
}

// ---------------------------------------------------------------------------
extern "C" void kernel_launch(void* const* d_in, const int* in_sizes, int n_in,
                              void* d_out, int out_size, void* d_ws, size_t ws_size,
                              hipStream_t stream) {
  const float* coord = (const float*)d_in[0];
  const float* feat  = (const float*)d_in[1];
  const int*   batch = (const int*)d_in[2];
  // per layer l: Wl, bl, Wr, br, att, bias at d_in[3 + 6*l .. ]
  const float* Wl[3]   = {(const float*)d_in[3],  (const float*)d_in[9],  (const float*)d_in[15]};
  const float* bl[3]   = {(const float*)d_in[4],  (const float*)d_in[10], (const float*)d_in[16]};
  const float* Wr[3]   = {(const float*)d_in[5],  (const float*)d_in[11], (const float*)d_in[17]};
  const float* br[3]   = {(const float*)d_in[6],  (const float*)d_in[12], (const float*)d_in[18]};
  const float* attL[3] = {(const float*)d_in[7],  (const float*)d_in[13], (const float*)d_in[19]};
  const float* bsL[3]  = {(const float*)d_in[8],  (const float*)d_in[14], (const float*)d_in[20]};
  const float* Wf = (const float*)d_in[21];
  const float* bf = (const float*)d_in[22];
  float* outp = (float*)d_out;

  // Workspace partition (~36.5 MB total)
  char*  ws  = (char*)d_ws;
  size_t off = 0;
  auto alloc = [&](size_t bytes) -> void* {
    void* p = ws + off;
    off = (off + bytes + 255) & ~(size_t)255;
    return p;
  };
  float* cand_d = (float*)alloc((size_t)NTOT * SPLIT * KNBR * sizeof(float));
  int*   cand_j = (int*)  alloc((size_t)NTOT * SPLIT * KNBR * sizeof(int));
  int*   idx    = (int*)  alloc((size_t)NTOT * KNBR * sizeof(int));
  float* xlb    = (float*)alloc((size_t)NTOT * HIDC * sizeof(float));
  float* xrb    = (float*)alloc((size_t)NTOT * HIDC * sizeof(float));
  float* x0     = (float*)alloc((size_t)NTOT * HIDC * sizeof(float));
  float* x1     = (float*)alloc((size_t)NTOT * HIDC * sizeof(float));
  (void)ws_size; (void)n_in; (void)in_sizes; (void)out_size;

  // kNN graph construction
  knn_pass1_kernel<<<dim3(NTOT / TILE, SPLIT), TILE, 0, stream>>>(coord, batch, cand_d, cand_j);
  knn_merge_kernel<<<NTOT / 256, 256, 0, stream>>>(cand_d, cand_j, idx);

  // Layer 0 (din = 6)
  gemm_bias_wmma_kernel<6><<<NTOT / 16, 256, 0, stream>>>(feat, Wl[0], bl[0], xlb);
  gemm_bias_wmma_kernel<6><<<NTOT / 16, 256, 0, stream>>>(feat, Wr[0], br[0], xrb);
  gat_aggregate_kernel<<<NTOT / 8, 256, 0, stream>>>(xlb, xrb, idx, attL[0], bsL[0], x0);

  // Layer 1 (din = 128)
  gemm_bias_wmma_kernel<HIDC><<<NTOT / 16, 256, 0, stream>>>(x0, Wl[1], bl[1], xlb);
  gemm_bias_wmma_kernel<HIDC><<<NTOT / 16, 256, 0, stream>>>(x0, Wr[1], br[1], xrb);
  gat_aggregate_kernel<<<NTOT / 8, 256, 0, stream>>>(xlb, xrb, idx, attL[1], bsL[1], x1);

  // Layer 2 (din = 128)
  gemm_bias_wmma_kernel<HIDC><<<NTOT / 16, 256, 0, stream>>>(x1, Wl[2], bl[2], xlb);
  gemm_bias_wmma_kernel<HIDC><<<NTOT / 16, 256, 0, stream>>>(x1, Wr[2], br[2], xrb);
  gat_aggregate_kernel<<<NTOT / 8, 256, 0, stream>>>(xlb, xrb, idx, attL[2], bsL[2], x0);

  // Classifier head
  final_linear_kernel<<<NTOT / 256, 256, 0, stream>>>(x0, Wf, bf, outp);
}